// NoisyTopkRouter_45715631898864
// MI455X (gfx1250) — compile-verified
//
#include <hip/hip_runtime.h>
#include <math.h>

typedef __attribute__((ext_vector_type(2))) float v2f;
typedef __attribute__((ext_vector_type(8))) float v8f;

#define DIM             2048
#define NE              8
#define NTOK            16384       // B*S = 4*4096
#define TOKS_PER_BLOCK  128
#define WAVES_PER_BLOCK 8
#define OUT_PROBS_N     (NTOK * NE)        // 131072
#define OUT_IDX_N       (NTOK * 2)         // 32768

// ---------------------------------------------------------------------------
// Fused router kernel: per 16-token tile, one fp32 WMMA chain computes all
// 16 logits (8 gate | 8 noise), then per-token softplus / noisy top-2 /
// masked softmax, plus block-reduced per-expert sums for the LB loss.
// ---------------------------------------------------------------------------
__global__ __launch_bounds__(256) void router_wmma_kernel(
    const float* __restrict__ x,      // [NTOK, DIM]
    const float* __restrict__ noise,  // [NTOK, NE]
    const float* __restrict__ Wg,     // [NE, DIM]
    const float* __restrict__ bg,     // [NE]
    const float* __restrict__ Wn,     // [NE, DIM]
    const float* __restrict__ bn,     // [NE]
    float* __restrict__ out,          // probs | indices | lb_loss
    float* __restrict__ ws)           // [2*NE] global partial sums
{
    __shared__ float scores[WAVES_PER_BLOCK][16][17];  // [tile][token][col], padded
    __shared__ float bsum[2 * NE];                     // [0..7]=prob sums, [8..15]=count sums

    const int tid   = threadIdx.x;
    const int wid   = tid >> 5;        // wave id in block (tile id)
    const int lane  = tid & 31;
    const int khalf = lane >> 4;       // 0 -> K in {0,1}; 1 -> K in {2,3}
    const int mcol  = lane & 15;       // A row (token) / B col (gate|noise)

    if (tid < 2 * NE) bsum[tid] = 0.0f;

    // ---- WMMA GEMM: [16 tokens x 2048] x [2048 x 16] ----------------------
    const int tile_base = blockIdx.x * TOKS_PER_BLOCK + wid * 16;
    const float* arow = x + (size_t)(tile_base + mcol) * DIM + 2 * khalf;
    const float* brow = (mcol < NE ? Wg + (size_t)mcol * DIM
                                   : Wn + (size_t)(mcol - NE) * DIM) + 2 * khalf;

    v8f acc = {};
    for (int k = 0; k < DIM; k += 32) {
        __builtin_prefetch(arow + k + 256, 0, 0);   // global_prefetch_b8
        #pragma unroll
        for (int kk = 0; kk < 32; kk += 4) {
            v2f a = *(const v2f*)(arow + k + kk);
            v2f b = *(const v2f*)(brow + k + kk);
            // (neg_a, A, neg_b, B, c_mod, C, reuse_a, reuse_b)
            acc = __builtin_amdgcn_wmma_f32_16x16x4_f32(
                false, a, false, b, (short)0, acc, false, false);
        }
    }

    // C layout: VGPR j, lanes 0-15 -> M=j; lanes 16-31 -> M=8+j; N = lane&15
    #pragma unroll
    for (int j = 0; j < 8; ++j)
        scores[wid][j + 8 * khalf][mcol] = acc[j];

    __syncthreads();

    // ---- per-token routing: threads 0..127 each own one token -------------
    if (tid < TOKS_PER_BLOCK) {
        const int tw = tid >> 4, tl = tid & 15;
        const int token = blockIdx.x * TOKS_PER_BLOCK + tid;
        const float* sc = scores[tw][tl];

        float noisy[NE];
        #pragma unroll
        for (int e = 0; e < NE; ++e) {
            float gl = sc[e] + bg[e];
            float nl = sc[NE + e] + bn[e];
            // softplus(nl) = max(nl,0) + log1p(exp(-|nl|))   (overflow-safe)
            float sp = fmaxf(nl, 0.0f) + log1pf(expf(-fabsf(nl)));
            noisy[e] = fmaf(noise[(size_t)token * NE + e], sp, gl);
        }

        // top-2 with jax.lax.top_k tie-breaking (first index wins)
        int i1 = 0; float m1 = noisy[0];
        #pragma unroll
        for (int e = 1; e < NE; ++e)
            if (noisy[e] > m1) { m1 = noisy[e]; i1 = e; }
        int i2 = (i1 == 0) ? 1 : 0; float m2 = noisy[i2];
        #pragma unroll
        for (int e = 0; e < NE; ++e)
            if (e != i1 && noisy[e] > m2) { m2 = noisy[e]; i2 = e; }

        // masked softmax over the two selected slots
        float e2  = expf(m2 - m1);
        float inv = 1.0f / (1.0f + e2);
        float p1 = inv, p2 = e2 * inv;

        #pragma unroll
        for (int e = 0; e < NE; ++e)
            out[(size_t)token * NE + e] = 0.0f;
        out[(size_t)token * NE + i1] = p1;
        out[(size_t)token * NE + i2] = p2;

        out[(size_t)OUT_PROBS_N + (size_t)token * 2 + 0] = (float)i1;
        out[(size_t)OUT_PROBS_N + (size_t)token * 2 + 1] = (float)i2;

        // LB-loss partials: prob sums and selection-count sums (ds_add_f32)
        atomicAdd(&bsum[i1], p1);
        atomicAdd(&bsum[i2], p2);
        atomicAdd(&bsum[NE + i1], 1.0f);
        atomicAdd(&bsum[NE + i2], 1.0f);
    }

    __syncthreads();
    if (tid < 2 * NE) atomicAdd(&ws[tid], bsum[tid]);  // global_atomic_add_f32
}

// ---------------------------------------------------------------------------
// lb_loss = NE * sum_e mean(probs_e) * mean(count_e)
// ---------------------------------------------------------------------------
__global__ void router_finalize_kernel(const float* __restrict__ ws,
                                       float* __restrict__ out)
{
    if (threadIdx.x == 0) {
        const float invN = 1.0f / (float)NTOK;
        float lb = 0.0f;
        #pragma unroll
        for (int e = 0; e < NE; ++e)
            lb += (ws[e] * invN) * (ws[NE + e] * invN);
        out[(size_t)OUT_PROBS_N + OUT_IDX_N] = (float)NE * lb;
    }
}

extern "C" void kernel_launch(void* const* d_in, const int* in_sizes, int n_in,
                              void* d_out, int out_size, void* d_ws, size_t ws_size,
                              hipStream_t stream) {
    const float* x     = (const float*)d_in[0];
    const float* noise = (const float*)d_in[1];
    const float* Wg    = (const float*)d_in[2];
    const float* bg    = (const float*)d_in[3];
    const float* Wn    = (const float*)d_in[4];
    const float* bn    = (const float*)d_in[5];
    float* out = (float*)d_out;
    float* ws  = (float*)d_ws;

    hipMemsetAsync(ws, 0, 2 * NE * sizeof(float), stream);
    router_wmma_kernel<<<NTOK / TOKS_PER_BLOCK, 256, 0, stream>>>(
        x, noise, Wg, bg, Wn, bn, out, ws);
    router_finalize_kernel<<<1, 32, 0, stream>>>(ws, out);
}